// PEERLayer_76355928588577
// MI455X (gfx1250) — compile-verified
//
#include <hip/hip_runtime.h>
#include <math.h>

// ---------------- constants (from reference) ----------------
#define BB   2
#define SS   128
#define DD   384
#define HH   6
#define HD   64
#define EHD  128
#define KK   16
#define SUBD 32
#define SQN  128
#define ROWS (BB * SS)          // 256 token rows
#define EXP_ROW (HD * EHD)      // 8192 floats per expert row (32 KB)

typedef __attribute__((ext_vector_type(2))) float v2f;
typedef __attribute__((ext_vector_type(8))) float v8f;
typedef __attribute__((ext_vector_type(4))) unsigned int v4u;
typedef __attribute__((ext_vector_type(8))) int v8i;
typedef __attribute__((ext_vector_type(4))) int v4i;

// ---------------------------------------------------------------------------
// Kernel A: fp32 WMMA GEMM  C[M,N] = A[M,K] * B[K,N]
// One wave (32 threads) per 16x16 output tile, v_wmma_f32_16x16x4_f32.
// ---------------------------------------------------------------------------
__global__ void gemm_wmma_f32(const float* __restrict__ A,
                              const float* __restrict__ B,
                              float* __restrict__ C,
                              int M, int N, int Kd) {
    const int lane = threadIdx.x;          // 0..31 (wave32, EXEC all ones)
    const int tn   = blockIdx.x * 16;
    const int tm   = blockIdx.y * 16;
    const int half = lane >> 4;            // 0 or 1
    const int l    = lane & 15;

    v8f acc = {};
    for (int k0 = 0; k0 < Kd; k0 += 4) {
        const int ka = k0 + half * 2;
        v2f a, b;
        a.x = A[(tm + l) * Kd + ka];
        a.y = A[(tm + l) * Kd + ka + 1];
        b.x = B[(ka) * N + tn + l];
        b.y = B[(ka + 1) * N + tn + l];
        acc = __builtin_amdgcn_wmma_f32_16x16x4_f32(
            false, a, false, b, (short)0, acc, false, false);
    }
#pragma unroll
    for (int v = 0; v < 8; ++v)
        C[(tm + half * 8 + v) * N + tn + l] = acc[v];
}

// ---------------------------------------------------------------------------
// TDM: DMA one contiguous 8192-float expert row from global into LDS.
// D# group0: count=1 | lds_addr | 57-bit global_addr | type=2 ("image")
// D# group1: data_size=4B, tensor_dim0=tile_dim0=8192, dim1=1, stride0=8192
// Groups 2/3 zero (<=2D tensor). Completion tracked with TENSORcnt.
// This toolchain's builtin is the 6-arg (clang-23 / therock-10.0) form.
// ---------------------------------------------------------------------------
__device__ inline void tdm_load_row(const float* gsrc, unsigned lds_byte_off) {
    unsigned long long ga = (unsigned long long)(uintptr_t)gsrc;
    v4u g0;
    g0[0] = 1u;                                            // count=1, user D#
    g0[1] = lds_byte_off;                                  // lds_addr
    g0[2] = (unsigned)(ga & 0xffffffffu);                  // global_addr[31:0]
    g0[3] = (unsigned)((ga >> 32) & 0x01ffffffu)           // global_addr[56:32]
          | 0x80000000u;                                   // type=2 in [127:126]
    v8i g1;
    g1[0] = 0x00020000;   // data_size=2 (4 bytes), no multicast/pad/iterate
    g1[1] = 0x20000000;   // tensor_dim0[15:0]=8192 in bits[31:16]
    g1[2] = 0x00010000;   // tensor_dim0 hi=0 ; tensor_dim1=1 in bits[31:16]
    g1[3] = 0x20000000;   // tensor_dim1 hi=0 ; tile_dim0=8192 in bits[31:16]
    g1[4] = 0x00000001;   // tile_dim1=1, tile_dim2=0
    g1[5] = 8192;         // tensor_dim0_stride[31:0]
    g1[6] = 0;            // stride hi, tensor_dim1_stride lo
    g1[7] = 0;
    v4i z4 = {0, 0, 0, 0};
    v8i z8 = {0, 0, 0, 0, 0, 0, 0, 0};
    __builtin_amdgcn_tensor_load_to_lds(g0, g1, z4, z4, z8, 0);
}

// ---------------------------------------------------------------------------
// Bitonic sort (descending) of 128 (key,idx) pairs in LDS, 64 worker lanes.
// ---------------------------------------------------------------------------
__device__ inline void bitonic128_desc(float* key, int* idx, int t) {
    for (int k = 2; k <= 128; k <<= 1) {
        for (int j = k >> 1; j > 0; j >>= 1) {
            __syncthreads();
            if (t < 64) {
                int i = ((t & ~(j - 1)) << 1) | (t & (j - 1));
                int p = i | j;
                bool up = (i & k) == 0;           // descending block
                float ki = key[i], kp = key[p];
                bool sw = up ? (ki < kp) : (ki > kp);
                if (sw) {
                    key[i] = kp; key[p] = ki;
                    int tt = idx[i]; idx[i] = idx[p]; idx[p] = tt;
                }
            }
        }
    }
    __syncthreads();
}

// ---------------------------------------------------------------------------
// Kernel B: fused per-(token,head) PEER pipeline.
// grid = 1536 blocks, block = 128 threads (4 wave32s).
// Dynamic LDS: double buffer for wd rows, 2 * 8192 floats (64 KB).
// ---------------------------------------------------------------------------
__global__ void peer_expert_kernel(const float* __restrict__ x,
                                   const float* __restrict__ Q,
                                   const float* __restrict__ gamma,
                                   const float* __restrict__ beta,
                                   const float* __restrict__ ckeys,
                                   const float* __restrict__ cpkeys,
                                   const float* __restrict__ wdown,
                                   const float* __restrict__ wup,
                                   float* __restrict__ OH) {
    extern __shared__ float wdbuf[];       // [2][EXP_ROW]

    const int inst = blockIdx.x;           // 0..1535
    const int h    = inst % HH;
    const int row  = inst / HH;            // token row 0..255
    const int t    = threadIdx.x;          // 0..127

    __shared__ float qn[HD];
    __shared__ float xh[HD];
    __shared__ float key[128];
    __shared__ int   kidx[128];
    __shared__ float t1v[16]; __shared__ int t1i[16];
    __shared__ float t2v[16]; __shared__ int t2i[16];
    __shared__ float jv[256];
    __shared__ float rv[128]; __shared__ int ri[128];
    __shared__ float fsc[KK];
    __shared__ int   eidx[KK];
    __shared__ float rw[KK];
    __shared__ float hid[EHD];
    __shared__ float red[128];

    // ---------------- LayerNorm of q head ----------------
    float qv = 0.f;
    if (t < HD) qv = Q[row * DD + h * HD + t];
    rv[t] = (t < HD) ? qv : 0.f;
    __syncthreads();
    for (int off = 64; off > 0; off >>= 1) {
        if (t < off) rv[t] += rv[t + off];
        __syncthreads();
    }
    const float mean = rv[0] * (1.0f / HD);
    __syncthreads();
    rv[t] = (t < HD) ? (qv - mean) * (qv - mean) : 0.f;
    __syncthreads();
    for (int off = 64; off > 0; off >>= 1) {
        if (t < off) rv[t] += rv[t + off];
        __syncthreads();
    }
    const float var = rv[0] * (1.0f / HD);
    __syncthreads();
    if (t < HD) {
        qn[t] = (qv - mean) * rsqrtf(var + 1e-5f) * gamma[t] + beta[t];
        xh[t] = x[row * DD + h * HD + t];      // original x head for experts
    }
    __syncthreads();

    // ---------------- product-key scores ----------------
    float s1 = 0.f, s2 = 0.f;
    {
        const float* ck  = ckeys  + t * SUBD;
        const float* ck2 = cpkeys + t * SUBD;
#pragma unroll 8
        for (int d = 0; d < SUBD; ++d) {
            s1 += qn[d]        * ck[d];
            s2 += qn[SUBD + d] * ck2[d];
        }
    }

    // top-16 of each score list (the joint top-16 can only come from the
    // 16x16 corner of the sorted joint matrix, so k_pre=64 collapses to 16)
    key[t] = s1; kidx[t] = t;
    bitonic128_desc(key, kidx, t);
    if (t < 16) { t1v[t] = key[t]; t1i[t] = kidx[t]; }
    __syncthreads();

    key[t] = s2; kidx[t] = t;
    bitonic128_desc(key, kidx, t);
    if (t < 16) { t2v[t] = key[t]; t2i[t] = kidx[t]; }
    __syncthreads();

    // ---------------- joint top-16 over 16x16 candidates ----------------
    jv[t]       = t1v[t >> 4]       + t2v[t & 15];
    jv[t + 128] = t1v[(t >> 4) + 8] + t2v[t & 15];
    __syncthreads();

    for (int r = 0; r < KK; ++r) {
        float v0 = jv[t], v1 = jv[t + 128];
        float bv; int bi;
        if (v1 > v0) { bv = v1; bi = t + 128; } else { bv = v0; bi = t; }
        rv[t] = bv; ri[t] = bi;
        __syncthreads();
        for (int off = 64; off > 0; off >>= 1) {
            if (t < off) {
                if (rv[t + off] > rv[t] ||
                    (rv[t + off] == rv[t] && ri[t + off] < ri[t])) {
                    rv[t] = rv[t + off]; ri[t] = ri[t + off];
                }
            }
            __syncthreads();
        }
        if (t == 0) {
            int c = ri[0];
            fsc[r]  = rv[0];
            eidx[r] = t1i[c >> 4] * SQN + t2i[c & 15];
            jv[c]   = -3.402823e38f;
        }
        __syncthreads();
    }

    // ---------------- routing softmax (scores already descending) --------
    if (t == 0) {
        float m = fsc[0], sum = 0.f;
        for (int k = 0; k < KK; ++k) { float e = __expf(fsc[k] - m); rw[k] = e; sum += e; }
        float inv = 1.0f / sum;
        for (int k = 0; k < KK; ++k) rw[k] *= inv;
    }
    __syncthreads();

    // ---------------- expert loop: TDM double-buffered wd streaming ------
    const unsigned wd_lds0 = (unsigned)(uintptr_t)&wdbuf[0];

    // preload expert 0's wd row into buffer 0 (wave 0 issues; EXEC ignored)
    if (t < 32) {
        tdm_load_row(wdown + (size_t)eidx[0] * EXP_ROW, wd_lds0);
    }

    float acc = 0.f;                       // output dim d = t (t < 64)
    for (int k = 0; k < KK; ++k) {
        const float* wu = wup + (size_t)eidx[k] * EXP_ROW;

        // wave 0: kick off DMA of next expert's wd row, then wait for wd_k.
        // TDM ops complete in issue order, so tensorcnt<=1 means wd_k landed
        // while wd_{k+1} is still streaming (full overlap with compute).
        if (t < 32) {
            if (k + 1 < KK) {
                tdm_load_row(wdown + (size_t)eidx[k + 1] * EXP_ROW,
                             wd_lds0 + (unsigned)(((k + 1) & 1) * EXP_ROW * 4));
                __builtin_amdgcn_s_wait_tensorcnt(1);
            } else {
                __builtin_amdgcn_s_wait_tensorcnt(0);
            }
        }
        __syncthreads();                   // wd_k visible to all waves

        if (k + 1 < KK) {                  // prefetch next expert's wu row
            __builtin_prefetch(wup + (size_t)eidx[k + 1] * EXP_ROW + t * 64, 0, 1);
        }

        // hidden[e=t] = sum_d xh[d] * wd[d*128 + e]   (from LDS)
        const float* wdl = wdbuf + (k & 1) * EXP_ROW;
        float hsum = 0.f;
#pragma unroll 8
        for (int d = 0; d < HD; ++d) hsum += xh[d] * wdl[d * EHD + t];

        // exact GELU, then routing weight
        float g = 0.5f * hsum * (1.0f + erff(hsum * 0.70710678118654752f));
        hid[t] = g * rw[k];
        __syncthreads();

        // out[d] += sum_e hid[e] * wu[e*64 + d]  (split e-range over 2 halves)
        const int d  = t & 63;
        const int hf = t >> 6;
        const float* wuh = wu + hf * 64 * HD;
        const float* hh  = hid + hf * 64;
        float p = 0.f;
#pragma unroll 8
        for (int e = 0; e < 64; ++e) p += hh[e] * wuh[e * HD + d];
        red[t] = p;
        __syncthreads();
        if (t < HD) acc += red[t] + red[t + 64];
        __syncthreads();                   // protect hid/red before next k
    }
    if (t < HD) OH[row * DD + h * HD + t] = acc;
}

// ---------------------------------------------------------------------------
// Launch: GEMM(q) -> fused PEER (TDM-pipelined experts) -> GEMM(out)
// ---------------------------------------------------------------------------
extern "C" void kernel_launch(void* const* d_in, const int* in_sizes, int n_in,
                              void* d_out, int out_size, void* d_ws, size_t ws_size,
                              hipStream_t stream) {
    const float* x      = (const float*)d_in[0];
    const float* Wq     = (const float*)d_in[1];
    const float* gamma  = (const float*)d_in[2];
    const float* beta   = (const float*)d_in[3];
    const float* ckeys  = (const float*)d_in[4];
    const float* cpkeys = (const float*)d_in[5];
    const float* wdown  = (const float*)d_in[6];
    const float* wup    = (const float*)d_in[7];
    const float* Wout   = (const float*)d_in[8];
    float* out = (float*)d_out;

    float* Qbuf  = (float*)d_ws;               // ROWS*DD floats
    float* OHbuf = Qbuf + (size_t)ROWS * DD;   // ROWS*DD floats

    // q = x @ W_q  (256x384 @ 384x384), fp32 WMMA
    gemm_wmma_f32<<<dim3(DD / 16, ROWS / 16), 32, 0, stream>>>(x, Wq, Qbuf, ROWS, DD, DD);

    // fused LN + product-key routing + TDM-pipelined expert MoE
    const size_t dyn_lds = (size_t)2 * EXP_ROW * sizeof(float);   // 64 KB
    peer_expert_kernel<<<ROWS * HH, 128, dyn_lds, stream>>>(
        x, Qbuf, gamma, beta, ckeys, cpkeys, wdown, wup, OHbuf);

    // out = out_heads @ W_out
    gemm_wmma_f32<<<dim3(DD / 16, ROWS / 16), 32, 0, stream>>>(OHbuf, Wout, out, ROWS, DD, DD);
}